// Resample_45561013076303
// MI455X (gfx1250) — compile-verified
//
#include <hip/hip_runtime.h>

// Bilinear splat (resample_to_map) for MI455X / gfx1250.
// This op is a random-target scatter-add: atomic/L2-bound, zero matmul
// structure (per-pixel outer product, no contraction dim) -> WMMA is
// inapplicable by construction. Optimization targets instead:
//   * wave32 blocks, coalesced b64/b128 vector memory
//   * weights/addresses computed once per pixel, reused for 32 channels
//   * no-return global_atomic_add_f32 (hardware f32 atomics at L2)
//   * global_prefetch_b8 of the next 8MB-strided channel slice

namespace {
constexpr int B  = 2;
constexpr int C  = 16;
constexpr int H  = 1024;
constexpr int W  = 2048;
constexpr int OH = 1024;
constexpr int OW = 2048;
constexpr int HW  = H * W;    // 2,097,152 input pixels
constexpr int OHW = OH * OW;  // 2,097,152 output pixels per plane
constexpr int BC  = B * C;    // 32 channel planes
}  // namespace

__global__ __launch_bounds__(256) void zero_out_kernel(float4* __restrict__ out,
                                                       int n4) {
  const int i = blockIdx.x * blockDim.x + threadIdx.x;
  if (i < n4) {
    out[i] = make_float4(0.f, 0.f, 0.f, 0.f);  // global_store_b128
  }
}

__global__ __launch_bounds__(256) void splat_kernel(
    const float* __restrict__ x,        // [B,C,H,W]
    const float2* __restrict__ smap,    // [H,W] of (px,py)
    float* __restrict__ out) {          // [B,C,OH,OW], pre-zeroed
  const int pix = blockIdx.x * blockDim.x + threadIdx.x;
  if (pix >= HW) return;

  // Coalesced 8-byte load of this pixel's target coordinate.
  const float2 p = smap[pix];

  const float x0f = floorf(p.x);
  const float y0f = floorf(p.y);
  const float dx  = p.x - x0f;
  const float dy  = p.y - y0f;
  const int x0 = (int)x0f;
  const int y0 = (int)y0f;
  const int x1 = x0 + 1;
  const int y1 = y0 + 1;

  // Reference semantics: weight *= valid; fold the mask into the weight so
  // the 32-channel hot loop is branch-free.
  const float vx0 = (x0 >= 0 && x0 < OW) ? 1.f : 0.f;
  const float vx1 = (x1 >= 0 && x1 < OW) ? 1.f : 0.f;
  const float vy0 = (y0 >= 0 && y0 < OH) ? 1.f : 0.f;
  const float vy1 = (y1 >= 0 && y1 < OH) ? 1.f : 0.f;

  const float w00 = (1.f - dx) * (1.f - dy) * (vx0 * vy0);
  const float w10 = dx * (1.f - dy) * (vx1 * vy0);
  const float w01 = (1.f - dx) * dy * (vx0 * vy1);
  const float w11 = dx * dy * (vx1 * vy1);

  const int xc0 = min(max(x0, 0), OW - 1);
  const int xc1 = min(max(x1, 0), OW - 1);
  const int yc0 = min(max(y0, 0), OH - 1);
  const int yc1 = min(max(y1, 0), OH - 1);

  const int f00 = yc0 * OW + xc0;
  const int f10 = yc0 * OW + xc1;
  const int f01 = yc1 * OW + xc0;
  const int f11 = yc1 * OW + xc1;

  const float* xp = x + pix;

#pragma unroll 4
  for (int bc = 0; bc < BC; ++bc) {
    if (bc + 1 < BC) {
      // Next channel plane is 8 MB away: hide its latency behind the four
      // in-flight atomics of this iteration (lowers to global_prefetch_b8).
      __builtin_prefetch(xp + (size_t)(bc + 1) * HW, /*rw=*/0, /*loc=*/1);
    }
    const float v = xp[(size_t)bc * HW];          // coalesced across lanes
    float* o = out + (size_t)bc * OHW;
    // Result unused -> compiler emits non-returning global_atomic_add_f32.
    atomicAdd(o + f00, v * w00);
    atomicAdd(o + f10, v * w10);
    atomicAdd(o + f01, v * w01);
    atomicAdd(o + f11, v * w11);
  }
}

extern "C" void kernel_launch(void* const* d_in, const int* in_sizes, int n_in,
                              void* d_out, int out_size, void* d_ws,
                              size_t ws_size, hipStream_t stream) {
  const float*  x    = (const float*)d_in[0];
  const float2* smap = (const float2*)d_in[1];
  float*        out  = (float*)d_out;

  // 1) Zero the accumulator (required every call: harness poisons d_out and
  //    atomics accumulate).
  const int n4 = out_size / 4;  // 64M floats -> 16M float4 stores
  zero_out_kernel<<<(n4 + 255) / 256, 256, 0, stream>>>((float4*)out, n4);

  // 2) Scatter-add: one thread per input pixel, 32 channel planes each.
  splat_kernel<<<(HW + 255) / 256, 256, 0, stream>>>(x, smap, out);
}